// GyroLoss_82282983457053
// MI455X (gfx1250) — compile-verified
//
#include <hip/hip_runtime.h>
#include <math.h>

#define HUBER_INV 200.0f   // 1/0.005
#define N0 5

typedef __attribute__((ext_vector_type(2))) float v2f;
typedef __attribute__((ext_vector_type(8))) float v8f;

// Rodrigues: rotation vector -> 3x3 rotation matrix (row-major R[9]).
// Matches reference: small-angle branch at t2 < 1e-8.
__device__ __forceinline__ void so3_exp_dev(float x, float y, float z, float R[9]) {
    float t2 = x * x + y * y + z * z;
    bool small = t2 < 1e-8f;
    float safe = small ? 1.0f : t2;
    float theta = sqrtf(safe);
    float s = sinf(theta);
    float c = cosf(theta);
    float A = small ? (1.0f - t2 * (1.0f / 6.0f)) : (s / theta);
    float B = small ? (0.5f - t2 * (1.0f / 24.0f)) : ((1.0f - c) / safe);
    // R = I + A*K + B*K^2,  K = skew(x,y,z)
    R[0] = 1.0f - B * (y * y + z * z);
    R[1] = -A * z + B * (x * y);
    R[2] =  A * y + B * (x * z);
    R[3] =  A * z + B * (x * y);
    R[4] = 1.0f - B * (x * x + z * z);
    R[5] = -A * x + B * (y * z);
    R[6] = -A * y + B * (x * z);
    R[7] =  A * x + B * (y * z);
    R[8] = 1.0f - B * (x * x + y * y);
}

// out = so3_log(R^T * H)   (both row-major 3x3)
__device__ __forceinline__ void rtm_log(const float R[9], const float H[9], float out[3]) {
    float M[9];
#pragma unroll
    for (int i = 0; i < 3; ++i)
#pragma unroll
        for (int j = 0; j < 3; ++j)
            M[i * 3 + j] = R[0 * 3 + i] * H[0 * 3 + j]
                         + R[1 * 3 + i] * H[1 * 3 + j]
                         + R[2 * 3 + i] * H[2 * 3 + j];
    float tr  = M[0] + M[4] + M[8];
    float cth = fminf(fmaxf(0.5f * (tr - 1.0f), -1.0f), 1.0f);
    bool small = cth > 1.0f - 1e-6f;
    float sc  = small ? 0.0f : cth;
    float ang = acosf(sc);
    float f   = small ? 0.5f : ang / (2.0f * sinf(ang));
    out[0] = f * (M[7] - M[5]);
    out[1] = f * (M[2] - M[6]);
    out[2] = f * (M[3] - M[1]);
}

__device__ __forceinline__ float huber_u(float r) {
    float z  = r * HUBER_INV;
    float az = fabsf(z);
    return az < 1.0f ? 0.5f * z * z : az - 0.5f;
}

__global__ void __launch_bounds__(256)
gyro_partial(const float* __restrict__ xs, const float* __restrict__ hat,
             float* __restrict__ partial, int T, int rows) {
    int r = blockIdx.x * blockDim.x + threadIdx.x;
    // Clamp instead of branching so EXEC stays all-1s through the WMMA reduce.
    int rr = r < rows ? r : rows - 1;
    float valid = (r < rows) ? 1.0f : 0.0f;
    int t = rr % T;
    float tmask = (t >= N0) ? 1.0f : 0.0f;

    const float* xp = xs  + (size_t)rr * 9;
    const float* hp = hat + (size_t)rr * 15;

    float wx = xp[0], wy = xp[1], wz = xp[2];
    float dvx = xp[3], dvy = xp[4], dvz = xp[5];
    float dpx = xp[6], dpy = xp[7], dpz = xp[8];

    float R[9], H[9], Xi[9];
    so3_exp_dev(wx, wy, wz, R);
    so3_exp_dev(hp[0], hp[1], hp[2], H);
    so3_exp_dev(hp[6], hp[7], hp[8], Xi);

    float r1[3], r3[3];
    rtm_log(R, H, r1);    // * 6 below
    rtm_log(R, Xi, r3);

    float acc = 0.0f;
    acc += huber_u(6.0f * r1[0]) + huber_u(6.0f * r1[1]) + huber_u(6.0f * r1[2]);
    acc += huber_u(6.0f * (dvx - hp[3])) + huber_u(6.0f * (dvy - hp[4])) + huber_u(6.0f * (dvz - hp[5]));
    acc += huber_u(r3[0]) + huber_u(r3[1]) + huber_u(r3[2]);
    acc += huber_u(dvx - hp[9])  + huber_u(dvy - hp[10]) + huber_u(dvz - hp[11]);
    acc += huber_u(dpx - hp[12]) + huber_u(dpy - hp[13]) + huber_u(dpz - hp[14]);
    acc *= valid * tmask;

    // ---- wave32 reduction via V_WMMA_F32_16X16X4_F32 ----
    // A(16x4): lane m holds {K0=acc, K1=0}, lane m+16 holds {K2=acc, K3=0}.
    // B(4x16) = ones  =>  D[m][n] = acc[m] + acc[m+16] for every column n.
    v2f a; a.x = acc;  a.y = 0.0f;
    v2f b; b.x = 1.0f; b.y = 1.0f;
    v8f c = {};
    c = __builtin_amdgcn_wmma_f32_16x16x4_f32(false, a, false, b, (short)0, c, false, false);
    float s = c[0] + c[1] + c[2] + c[3] + c[4] + c[5] + c[6] + c[7]; // rows 0-7 (lanes<16) / 8-15 (lanes>=16)
    s += __shfl_xor(s, 16);                                          // full 32-lane total in every lane

    __shared__ float lds[8];
    int lane = threadIdx.x & 31;
    int wave = threadIdx.x >> 5;
    if (lane == 0) lds[wave] = s;
    __syncthreads();
    if (threadIdx.x == 0) {
        float bsum = 0.0f;
#pragma unroll
        for (int i = 0; i < 8; ++i) bsum += lds[i];   // fixed order: deterministic
        partial[blockIdx.x] = bsum;
    }
}

__global__ void __launch_bounds__(256)
final_reduce(const float* __restrict__ partial, int n, float scale, float* __restrict__ out) {
    __shared__ float sh[256];
    float s = 0.0f;
    for (int i = threadIdx.x; i < n; i += 256) s += partial[i];   // fixed per-thread order
    sh[threadIdx.x] = s;
    __syncthreads();
    for (int off = 128; off > 0; off >>= 1) {                     // fixed tree: deterministic
        if (threadIdx.x < off) sh[threadIdx.x] += sh[threadIdx.x + off];
        __syncthreads();
    }
    if (threadIdx.x == 0) out[0] = sh[0] * scale;
}

extern "C" void kernel_launch(void* const* d_in, const int* in_sizes, int n_in,
                              void* d_out, int out_size, void* d_ws, size_t ws_size,
                              hipStream_t stream) {
    (void)n_in; (void)out_size; (void)ws_size;
    const float* xs  = (const float*)d_in[0];   // (N,T,9)  f32
    const float* hat = (const float*)d_in[1];   // (N,T,15) f32
    float* out = (float*)d_out;
    float* partial = (float*)d_ws;

    int rows = in_sizes[0] / 9;   // N*T
    int N = 64;
    int T = rows / N;
    int nblk = (rows + 255) / 256;

    // W * HUBER^2 / (N*(T-N0)*15)
    double denom = (double)N * (double)(T - N0) * 15.0;
    float scale = (float)(1000000.0 * 0.005 * 0.005 / denom);

    gyro_partial<<<nblk, 256, 0, stream>>>(xs, hat, partial, T, rows);
    final_reduce<<<1, 256, 0, stream>>>(partial, nblk, scale, out);
}